// RNN_51943334478458
// MI455X (gfx1250) — compile-verified
//
#include <hip/hip_runtime.h>
#include <math.h>

// Problem sizes (fixed by the reference)
#define B_   256
#define T_   200
#define DIN_ 512
#define U_   1024
#define P_   512

// Tiling
#define BM 32      // block tile rows (batch)
#define BN 64      // block tile cols
#define BK 32      // K stage
#define LDA_S 36   // padded LDS row stride for A tile (conflict-free column reads)
#define LDB_S 72   // padded LDS row stride for B tile (half-wave rows on disjoint banks)
#define A_TILE (BM * LDA_S)
#define B_TILE (BK * LDB_S)

typedef float v2f __attribute__((ext_vector_type(2)));
typedef float v8f __attribute__((ext_vector_type(8)));

__device__ __forceinline__ float sigf(float v) {
    return 1.0f / (1.0f + __expf(-v));
}

// Issue this thread's share of one K-stage: 1 A float4 + 2 B float4s,
// DMA'd straight into LDS via the CDNA5 async-to-LDS path (ASYNCcnt).
__device__ __forceinline__ void async_stage(
    const float* aSrc, const float* bSrc0, const float* bSrc1,
    unsigned asDst, unsigned bsDst0, unsigned bsDst1)
{
    asm volatile(
        "global_load_async_to_lds_b128 %0, %1, off\n\t"
        "global_load_async_to_lds_b128 %2, %3, off\n\t"
        "global_load_async_to_lds_b128 %4, %5, off"
        :
        : "v"(asDst), "v"(aSrc), "v"(bsDst0), "v"(bSrc0), "v"(bsDst1), "v"(bSrc1)
        : "memory");
}

// acc += A[rowBaseA.., :K] * B[:, colBaseB..] with v_wmma_f32_16x16x4_f32.
// Double-buffered LDS tiles, async-DMA staged, prefetch overlapped with compute.
__device__ __forceinline__ void gemm_acc(
    const float* __restrict__ Ag, long lda,
    const float* __restrict__ Bg, int ldb,
    int K,
    float* As, float* Bs,           // LDS: As[2][A_TILE], Bs[2][B_TILE]
    int tid, int rowBaseA, int colBaseB,
    v8f& acc)
{
    const int lane = tid & 31;
    const int wid  = tid >> 5;
    const int wm   = (wid >> 2) << 4;   // 0 or 16
    const int wn   = (wid & 3) << 4;    // 0..48
    const int half = lane >> 4;         // 0 or 1
    const int idx  = lane & 15;

    // cooperative-staging coordinates
    const int arow = tid >> 3;          // 0..31
    const int acol = (tid & 7) << 2;    // 0..28 (float4)
    const int brow = tid >> 4;          // 0..15
    const int bcol = (tid & 15) << 2;   // 0..60 (float4)

    // Per-thread global source pointers for stage 0
    const float* aPtr  = Ag + (long)(rowBaseA + arow) * lda + acol;
    const float* b0Ptr = Bg + (long)brow * ldb + (colBaseB + bcol);
    const float* b1Ptr = b0Ptr + 16L * ldb;
    const long bStep = (long)BK * ldb;

    // LDS destination byte addresses (low 32 bits of the generic pointer
    // are the LDS offset on gfx1250)
    unsigned asD[2], bs0D[2], bs1D[2];
    #pragma unroll
    for (int i = 0; i < 2; ++i) {
        asD[i]  = (unsigned)(size_t)(As + i * A_TILE + arow * LDA_S + acol);
        bs0D[i] = (unsigned)(size_t)(Bs + i * B_TILE + brow * LDB_S + bcol);
        bs1D[i] = (unsigned)(size_t)(Bs + i * B_TILE + (brow + 16) * LDB_S + bcol);
    }

    const int nStages = K / BK;
    async_stage(aPtr, b0Ptr, b1Ptr, asD[0], bs0D[0], bs1D[0]);

    for (int s = 0; s < nStages; ++s) {
        const int cur = s & 1;
        if (s + 1 < nStages) {
            async_stage(aPtr + (long)(s + 1) * BK,
                        b0Ptr + (long)(s + 1) * bStep,
                        b1Ptr + (long)(s + 1) * bStep,
                        asD[cur ^ 1], bs0D[cur ^ 1], bs1D[cur ^ 1]);
            // in-order completion: current stage (3 older ops) done when <=3 remain
            asm volatile("s_wait_asynccnt 0x3" ::: "memory");
        } else {
            asm volatile("s_wait_asynccnt 0x0" ::: "memory");
        }
        __syncthreads();

        const float* AsC = As + cur * A_TILE;
        const float* BsC = Bs + cur * B_TILE;
        #pragma unroll
        for (int kk = 0; kk < BK; kk += 4) {
            const int ka = kk + half * 2;
            v2f a, b;
            a.x = AsC[(wm + idx) * LDA_S + ka];
            a.y = AsC[(wm + idx) * LDA_S + ka + 1];
            b.x = BsC[(ka)     * LDB_S + wn + idx];
            b.y = BsC[(ka + 1) * LDB_S + wn + idx];
            acc = __builtin_amdgcn_wmma_f32_16x16x4_f32(
                false, a, false, b, (short)0, acc, false, false);
        }
        __syncthreads();
    }
}

// Step kernel 1: h_out = tanh( x_t @ Wk + h_in @ R + sp_in @ F + bias )
// (sp_in already holds sigmoid(p) — applied by the producer)
__global__ void __launch_bounds__(256)
rnn_step_h(const float* __restrict__ x_t,   // base = x + t*DIN, row stride T*DIN
           const float* __restrict__ Wk,    // [DIN, U]
           const float* __restrict__ R,     // [U, U]
           const float* __restrict__ F,     // [P, U]
           const float* __restrict__ bias,  // [U]
           const float* __restrict__ h_in,  // [B, U]
           const float* __restrict__ sp_in, // [B, P] = sigmoid(p_prev)
           float* __restrict__ h_out)       // [B, U]
{
    __shared__ float As[2 * A_TILE];
    __shared__ float Bs[2 * B_TILE];

    const int tid  = threadIdx.x;
    const int lane = tid & 31;
    const int wid  = tid >> 5;
    const int wm   = (wid >> 2) << 4;
    const int wn   = (wid & 3) << 4;
    const int half = lane >> 4;
    const int idx  = lane & 15;

    const int rowBase = blockIdx.y * BM;
    const int colBase = blockIdx.x * BN;
    const int ng = colBase + wn + idx;

    v8f acc;
    const float bv = bias[ng];
    #pragma unroll
    for (int v = 0; v < 8; ++v) acc[v] = bv;

    gemm_acc(x_t,   (long)T_ * DIN_, Wk, U_, DIN_, As, Bs, tid, rowBase, colBase, acc);
    gemm_acc(h_in,  (long)U_,        R,  U_, U_,   As, Bs, tid, rowBase, colBase, acc);
    gemm_acc(sp_in, (long)P_,        F,  U_, P_,   As, Bs, tid, rowBase, colBase, acc);

    const int mg0 = rowBase + wm + half * 8;
    #pragma unroll
    for (int v = 0; v < 8; ++v) {
        h_out[(long)(mg0 + v) * U_ + ng] = tanhf(acc[v]);
    }
}

// Step kernel 2: p_new = h @ Pk + pb ; out[:,t,:] = p_new ; sp = sigmoid(p_new)
__global__ void __launch_bounds__(256)
rnn_step_p(const float* __restrict__ h,     // [B, U]
           const float* __restrict__ Pk,    // [U, P]
           const float* __restrict__ pb,    // [P]
           float* __restrict__ sp,          // [B, P] sigmoid state for next step
           float* __restrict__ out,         // [B, T, P]
           int t)
{
    __shared__ float As[2 * A_TILE];
    __shared__ float Bs[2 * B_TILE];

    const int tid  = threadIdx.x;
    const int lane = tid & 31;
    const int wid  = tid >> 5;
    const int wm   = (wid >> 2) << 4;
    const int wn   = (wid & 3) << 4;
    const int half = lane >> 4;
    const int idx  = lane & 15;

    const int rowBase = blockIdx.y * BM;
    const int colBase = blockIdx.x * BN;
    const int ng = colBase + wn + idx;

    v8f acc;
    const float bv = pb[ng];
    #pragma unroll
    for (int v = 0; v < 8; ++v) acc[v] = bv;

    gemm_acc(h, (long)U_, Pk, P_, U_, As, Bs, tid, rowBase, colBase, acc);

    const int mg0 = rowBase + wm + half * 8;
    #pragma unroll
    for (int v = 0; v < 8; ++v) {
        const int m = mg0 + v;
        const float val = acc[v];
        out[((long)m * T_ + t) * P_ + ng] = val;   // reference output: pre-activation
        sp[(long)m * P_ + ng] = sigf(val);         // recurrence consumes sigmoid(p)
    }
}

// h0 = 0 ; sp0 = sigmoid(0) = 0.5
__global__ void init_state(float* h, int nh, float* sp, int np) {
    const int i = blockIdx.x * blockDim.x + threadIdx.x;
    if (i < nh) h[i] = 0.0f;
    if (i < np) sp[i] = 0.5f;
}

extern "C" void kernel_launch(void* const* d_in, const int* in_sizes, int n_in,
                              void* d_out, int out_size, void* d_ws, size_t ws_size,
                              hipStream_t stream) {
    const float* x    = (const float*)d_in[0];  // [B, T, DIN]
    const float* Wk   = (const float*)d_in[1];  // [DIN, U]
    const float* R    = (const float*)d_in[2];  // [U, U]
    const float* F    = (const float*)d_in[3];  // [P, U]
    const float* Pk   = (const float*)d_in[4];  // [U, P]
    const float* bias = (const float*)d_in[5];  // [U]
    const float* pb   = (const float*)d_in[6];  // [P]
    float* out = (float*)d_out;

    // Workspace: double-buffered h state + sigmoid(p) state
    float* hA = (float*)d_ws;
    float* hB = hA + (size_t)B_ * U_;
    float* sp = hB + (size_t)B_ * U_;

    {
        const int nh = B_ * U_;
        const int np = B_ * P_;
        const int n  = nh > np ? nh : np;
        init_state<<<(n + 255) / 256, 256, 0, stream>>>(hA, nh, sp, np);
    }

    const dim3 grid_h(U_ / BN, B_ / BM);  // 16 x 8
    const dim3 grid_p(P_ / BN, B_ / BM);  //  8 x 8

    for (int t = 0; t < T_; ++t) {
        const float* hin  = (t & 1) ? hB : hA;
        float*       hout = (t & 1) ? hA : hB;
        rnn_step_h<<<grid_h, 256, 0, stream>>>(x + (size_t)t * DIN_, Wk, R, F, bias,
                                               hin, sp, hout);
        rnn_step_p<<<grid_p, 256, 0, stream>>>(hout, Pk, pb, sp, out, t);
    }
}